// VideoImitation_45603962749554
// MI455X (gfx1250) — compile-verified
//
#include <hip/hip_runtime.h>
#include <hip/hip_bf16.h>

typedef float v2f __attribute__((ext_vector_type(2)));
typedef float v8f __attribute__((ext_vector_type(8)));

// Problem constants
#define BATCH   8
#define DMODEL  512
#define TT      36
#define HWD     84            // 7*12
#define THW     3024          // 36*84
#define FF      128
#define NHEAD   4
#define HC      32            // FF/NHEAD
#define DEMO_T  4
#define OBS_T   32
#define DP      336           // DEMO_T*HW
#define OP      2688          // OBS_T*HW
#define CATK    420           // DP + HW
#define CATKP   432           // padded to 27*16
#define INV_TEMP 0.04419417382415922f  // 1/sqrt(512)

// ---- full-precision CDNA5 WMMA: D = A(16x4,f32) * B(4x16,f32) + C(16x16,f32)
__device__ __forceinline__ v8f wmma4v(v2f a, v2f b, v8f c) {
    return __builtin_amdgcn_wmma_f32_16x16x4_f32(false, a, false, b, (short)0, c,
                                                 false, false);
}
__device__ __forceinline__ v8f wmma4(float a0, float a1, float b0, float b1, v8f c) {
    v2f a = {a0, a1};
    v2f b = {b0, b1};
    return wmma4v(a, b, c);
}

// =====================================================================
// Conv1x1 as GEMM:  Y[m, n] = sum_k W[m,k] * X[k, n]   (per batch b)
// X element: X[b*xbs + k*xrs + xcol + n];  Y: Y[b*ybs + m*yrs + ycol + n]
// EPI==1: Y = Yold + relu(acc)   (obs residual)
// One wave computes a 32x16 tile (2 M-subtiles sharing B fragments);
// A pairs are contiguous 8B-aligned -> global_load_b64.
// 8 waves per 256-thread block. M % 32 == 0, N % 16 == 0, K % 4 == 0.
// =====================================================================
template <int EPI>
__global__ void gemm_conv(const float* __restrict__ W, const float* __restrict__ X,
                          long xbs, int xrs, int xcol,
                          float* __restrict__ Y, long ybs, int yrs, int ycol,
                          int M, int N, int K)
{
    const int wave = threadIdx.x >> 5;
    const int lane = threadIdx.x & 31;
    const int lr = lane & 15, h = lane >> 4;
    const int Mt2 = M >> 5, Nt = N >> 4;
    const int tile = blockIdx.x * (blockDim.x >> 5) + wave;
    if (tile >= Mt2 * Nt) return;
    const int m0 = (tile / Nt) << 5;
    const int n0 = (tile % Nt) << 4;
    const int b = blockIdx.y;

    const float* Wr0 = W + (long)(m0 + lr) * K + 2 * h;        // A row m0+lr
    const float* Wr1 = W + (long)(m0 + 16 + lr) * K + 2 * h;   // A row m0+16+lr
    const float* Xb  = X + (long)b * xbs + xcol + n0 + lr + (long)(2 * h) * xrs;

    v8f acc0 = {}, acc1 = {};
#pragma unroll 4
    for (int k0 = 0; k0 < K; k0 += 4) {
        v2f a0 = *(const v2f*)(Wr0 + k0);                      // b64
        v2f a1 = *(const v2f*)(Wr1 + k0);                      // b64
        float b0 = Xb[(long)k0 * xrs];
        float b1 = Xb[(long)(k0 + 1) * xrs];
        v2f bb = {b0, b1};
        acc0 = wmma4v(a0, bb, acc0);
        acc1 = wmma4v(a1, bb, acc1);
    }

    float* Yp = Y + (long)b * ybs + ycol + n0 + lr;
#pragma unroll
    for (int r = 0; r < 8; ++r) {
        int m = m0 + 8 * h + r;
        float v = acc0[r];
        if (EPI == 1) v = Yp[(long)m * yrs] + fmaxf(v, 0.0f);
        Yp[(long)m * yrs] = v;
    }
#pragma unroll
    for (int r = 0; r < 8; ++r) {
        int m = m0 + 16 + 8 * h + r;
        float v = acc1[r];
        if (EPI == 1) v = Yp[(long)m * yrs] + fmaxf(v, 0.0f);
        Yp[(long)m * yrs] = v;
    }
}

// =====================================================================
// Demo attention, fused per (b, head, 16-key tile).
// S[i,j] = (dk[:,i] . dq[:,j]) / TEMP + mask;  P = softmax_j(S);
// dv_att[c,i] = sum_j dv[c,j] * P[i,j]
// block = 256 threads (8 waves). grid = (21, NHEAD, B)
// =====================================================================
__global__ void demo_attn(const float* __restrict__ dk, const float* __restrict__ dq,
                          const float* __restrict__ dv, float* __restrict__ dvatt)
{
    __shared__ float s[16][DP];
    const int i0 = blockIdx.x << 4;
    const int nh = blockIdx.y;
    const int b  = blockIdx.z;
    const int wave = threadIdx.x >> 5;
    const int lane = threadIdx.x & 31;
    const int lr = lane & 15, h = lane >> 4;

    const long base = ((long)b * FF + nh * HC) * DP;   // [c][pos], row stride DP
    const float* Kb = dk + base;
    const float* Qb = dq + base;
    const float* Vb = dv + base;

    // ---- phase 1: score tiles S[16 x 336] (K = 32 channels)
    for (int jt = wave; jt < DP / 16; jt += 8) {
        const int j0 = jt << 4;
        v8f acc = {};
#pragma unroll
        for (int k0 = 0; k0 < HC; k0 += 4) {
            float a0 = Kb[(k0 + 2 * h) * DP + i0 + lr];        // A^T: dk[c][i]
            float a1 = Kb[(k0 + 2 * h + 1) * DP + i0 + lr];
            float b0 = Qb[(k0 + 2 * h) * DP + j0 + lr];        // B:   dq[c][j]
            float b1 = Qb[(k0 + 2 * h + 1) * DP + j0 + lr];
            acc = wmma4(a0, a1, b0, b1, acc);
        }
#pragma unroll
        for (int r = 0; r < 8; ++r) {
            int m = 8 * h + r;
            int i = i0 + m, j = j0 + lr;
            float v = acc[r] * INV_TEMP;
            if (j / HWD > i / HWD) v = -__builtin_inff();      // causal (key-time >= query-time)
            s[m][j] = v;
        }
    }
    __syncthreads();

    // ---- phase 2: softmax over j (length 336) per key row
#pragma unroll
    for (int rr = 0; rr < 2; ++rr) {
        int r = wave * 2 + rr;
        float mx = -3.4e38f;
        for (int j = lane; j < DP; j += 32) mx = fmaxf(mx, s[r][j]);
        for (int o = 16; o; o >>= 1) mx = fmaxf(mx, __shfl_xor(mx, o, 32));
        float sum = 0.f;
        for (int j = lane; j < DP; j += 32) { float e = __expf(s[r][j] - mx); s[r][j] = e; sum += e; }
        for (int o = 16; o; o >>= 1) sum += __shfl_xor(sum, o, 32);
        float inv = 1.0f / sum;
        for (int j = lane; j < DP; j += 32) s[r][j] *= inv;
    }
    __syncthreads();

    // ---- phase 3: dv_att tile (M=32 channels, N=16 keys, K=336)
    if (wave < 2) {
        const int m0 = wave << 4;
        v8f acc = {};
        const float* Vr = Vb + (long)(m0 + lr) * DP + 2 * h;   // 8B aligned pairs
        const float* Sr = &s[lr][2 * h];
        for (int k0 = 0; k0 < DP; k0 += 4) {
            v2f a  = *(const v2f*)(Vr + k0);                   // b64: dv[c][j..j+1]
            v2f bb = *(const v2f*)(Sr + k0);                   // ds b64: P[i][j..j+1]
            acc = wmma4v(a, bb, acc);
        }
        float* Ob = dvatt + ((long)b * NHEAD + nh) * HC * DP;
#pragma unroll
        for (int r = 0; r < 8; ++r) {
            int c = m0 + 8 * h + r;
            Ob[c * DP + i0 + lr] = acc[r];
        }
    }
}

// =====================================================================
// Obs attention, fused per (b, t, head). 420 keys streamed in 16-chunks.
// S[kk,q] = (cat_k[:,kk] . oq[:,q])/TEMP; P = softmax_q(S);
// out[c,q] = sum_kk cat_v[c,kk] * P[kk,q]
// block = 384 threads (12 waves = 2 c-tiles x 6 q-tiles). grid=(32,4,8)
// =====================================================================
__global__ void obs_attn(const float* __restrict__ dk, const float* __restrict__ dvatt,
                         const float* __restrict__ oq, const float* __restrict__ ok,
                         const float* __restrict__ ov, float* __restrict__ ovatt)
{
    __shared__ float s[16][96];
    const int t  = blockIdx.x;
    const int nh = blockIdx.y;
    const int b  = blockIdx.z;
    const int wave = threadIdx.x >> 5;     // 0..11
    const int lane = threadIdx.x & 31;
    const int lr = lane & 15, h = lane >> 4;
    const int mi = wave / 6, ni = wave % 6;

    const float* Kd = dk    + ((long)b * FF + nh * HC) * DP;            // [c][kk<336]
    const float* Vd = dvatt + ((long)b * NHEAD + nh) * HC * DP;         // [c][kk<336]
    const long obase = ((long)b * FF + nh * HC) * OP + t * HWD;
    const float* Ko = ok + obase;                                       // [c][q]
    const float* Vo = ov + obase;
    const float* Qo = oq + obase;

    v8f oacc = {};

    for (int kk0 = 0; kk0 < CATKP; kk0 += 16) {
        // phase 1: S tile [16 keys x 96 queries] (waves 0..5)
        if (wave < 6) {
            const int q0 = wave << 4;
            v8f acc = {};
#pragma unroll
            for (int k0 = 0; k0 < HC; k0 += 4) {
                int c0 = k0 + 2 * h, c1 = c0 + 1;
                int kk = kk0 + lr;
                float a0 = (kk < DP) ? Kd[c0 * DP + kk] : ((kk < CATK) ? Ko[c0 * OP + (kk - DP)] : 0.f);
                float a1 = (kk < DP) ? Kd[c1 * DP + kk] : ((kk < CATK) ? Ko[c1 * OP + (kk - DP)] : 0.f);
                int q = q0 + lr;
                float b0 = (q < HWD) ? Qo[c0 * OP + q] : 0.f;
                float b1 = (q < HWD) ? Qo[c1 * OP + q] : 0.f;
                acc = wmma4(a0, a1, b0, b1, acc);
            }
#pragma unroll
            for (int r = 0; r < 8; ++r) s[8 * h + r][q0 + lr] = acc[r] * INV_TEMP;
        }
        __syncthreads();

        // phase 2: per-key softmax over the 84 queries
        for (int r = wave; r < 16; r += 12) {
            float mx = -3.4e38f;
            for (int q = lane; q < HWD; q += 32) mx = fmaxf(mx, s[r][q]);
            for (int o = 16; o; o >>= 1) mx = fmaxf(mx, __shfl_xor(mx, o, 32));
            float sum = 0.f;
            for (int q = lane; q < HWD; q += 32) { float e = __expf(s[r][q] - mx); s[r][q] = e; sum += e; }
            for (int o = 16; o; o >>= 1) sum += __shfl_xor(sum, o, 32);
            float inv = 1.0f / sum;
            for (int q = lane; q < HWD; q += 32) s[r][q] *= inv;
        }
        __syncthreads();

        // phase 3: out[c,q] += cat_v[c, kk-chunk] * P   (all 12 waves)
        {
            const int m0 = mi << 4, q0 = ni << 4;
            const int c = m0 + lr;
#pragma unroll
            for (int k0 = 0; k0 < 16; k0 += 4) {
                int kkA = kk0 + k0 + 2 * h;                    // even: pair never straddles DP
                v2f a;
                if (kkA < DP)        a = *(const v2f*)&Vd[c * DP + kkA];
                else if (kkA < CATK) a = *(const v2f*)&Vo[c * OP + (kkA - DP)];
                else                 { a[0] = 0.f; a[1] = 0.f; }
                float b0 = s[k0 + 2 * h][q0 + lr];
                float b1 = s[k0 + 2 * h + 1][q0 + lr];
                v2f bb = {b0, b1};
                oacc = wmma4v(a, bb, oacc);
            }
        }
        __syncthreads();
    }

    // store into W_o input layout: ovatt[b][nh*HC + c][t*84 + q]
    const int m0 = mi << 4, q0 = ni << 4;
    const int q = q0 + lr;
    if (q < HWD) {
        float* Ob = ovatt + ((long)b * FF + nh * HC) * OP + t * HWD + q;
#pragma unroll
        for (int r = 0; r < 8; ++r) {
            int c = m0 + 8 * h + r;
            Ob[(long)c * OP] = oacc[r];
        }
    }
}

// =====================================================================
// BatchNorm stats: mean & rstd per channel over (B, T, H, W)
// =====================================================================
__global__ void bn_stats(const float* __restrict__ x, float* __restrict__ meanv,
                         float* __restrict__ rstdv)
{
    const int c = blockIdx.x;
    const int tid = threadIdx.x;
    const int NEL = BATCH * THW;      // 24192
    double sum = 0.0, sq = 0.0;
    for (int f = tid; f < NEL; f += 256) {
        int b = f / THW, idx = f - b * THW;
        float v = x[((long)b * DMODEL + c) * THW + idx];
        sum += v; sq += (double)v * v;
    }
    __shared__ double rs[256], rs2[256];
    rs[tid] = sum; rs2[tid] = sq;
    __syncthreads();
    for (int o = 128; o; o >>= 1) {
        if (tid < o) { rs[tid] += rs[tid + o]; rs2[tid] += rs2[tid + o]; }
        __syncthreads();
    }
    if (tid == 0) {
        double m = rs[0] / NEL;
        double var = rs2[0] / NEL - m * m;
        meanv[c] = (float)m;
        rstdv[c] = rsqrtf((float)var + 1e-5f);
    }
}

__global__ void bn_apply(const float* __restrict__ xin, float* __restrict__ xout,
                         const float* __restrict__ meanv, const float* __restrict__ rstdv,
                         const float* __restrict__ gamma, const float* __restrict__ beta)
{
    long i = (long)blockIdx.x * 256 + threadIdx.x;
    const long TOT = (long)BATCH * DMODEL * THW;
    if (i >= TOT) return;
    int c = (int)((i / THW) % DMODEL);
    xout[i] = gamma[c] * (xin[i] - meanv[c]) * rstdv[c] + beta[c];
}

// =====================================================================
extern "C" void kernel_launch(void* const* d_in, const int* in_sizes, int n_in,
                              void* d_out, int out_size, void* d_ws, size_t ws_size,
                              hipStream_t stream)
{
    const float* in    = (const float*)d_in[0];
    const float* dwq   = (const float*)d_in[1];
    const float* dwk   = (const float*)d_in[2];
    const float* dwv   = (const float*)d_in[3];
    const float* dwo   = (const float*)d_in[4];
    const float* owq   = (const float*)d_in[5];
    const float* owk   = (const float*)d_in[6];
    const float* owv   = (const float*)d_in[7];
    const float* owo   = (const float*)d_in[8];
    const float* gamma = (const float*)d_in[9];
    const float* beta  = (const float*)d_in[10];

    const long XSZ = (long)BATCH * DMODEL * THW;   // 12,386,304
    const long DQS = (long)BATCH * FF * DP;        //    344,064
    const long OQS = (long)BATCH * FF * OP;        //  2,752,512

    float* ws    = (float*)d_ws;
    float* xbuf  = ws;                 ws += XSZ;
    float* dqb   = ws;                 ws += DQS;
    float* dkb   = ws;                 ws += DQS;
    float* dvb   = ws;                 ws += DQS;
    float* dvatt = ws;                 ws += DQS;
    float* oqb   = ws;                 ws += OQS;
    float* okb   = ws;                 ws += OQS;
    float* ovb   = ws;                 ws += OQS;
    float* ovatt = ws;                 ws += OQS;
    float* meanb = ws;                 ws += DMODEL;
    float* rstdb = ws;                 ws += DMODEL;

    hipMemcpyAsync(xbuf, in, XSZ * sizeof(float), hipMemcpyDeviceToDevice, stream);

    // tiles = (M/32)*(N/16); blocks = ceil(tiles/8)
    const int tb_demo = (4 * 21 + 7) / 8;     // M=128, N=336  -> 84 tiles  -> 11
    const int tb_obs  = (4 * 168 + 7) / 8;    // M=128, N=2688 -> 672 tiles -> 84
    const int tb_wo   = (16 * 168 + 7) / 8;   // M=512, N=2688 -> 2688 tiles-> 336

    for (int L = 0; L < 2; ++L) {
        const float* Wdq = dwq + (long)L * FF * DMODEL;
        const float* Wdk = dwk + (long)L * FF * DMODEL;
        const float* Wdv = dwv + (long)L * FF * DMODEL;
        const float* Woq = owq + (long)L * FF * DMODEL;
        const float* Wok = owk + (long)L * FF * DMODEL;
        const float* Wov = owv + (long)L * FF * DMODEL;
        const float* Woo = owo + (long)L * DMODEL * FF;

        // demo Q/K/V: M=128, N=336, K=512
        dim3 gd(tb_demo, BATCH);
        gemm_conv<0><<<gd, 256, 0, stream>>>(Wdq, xbuf, (long)DMODEL * THW, THW, 0,
                                             dqb, (long)FF * DP, DP, 0, FF, DP, DMODEL);
        gemm_conv<0><<<gd, 256, 0, stream>>>(Wdk, xbuf, (long)DMODEL * THW, THW, 0,
                                             dkb, (long)FF * DP, DP, 0, FF, DP, DMODEL);
        gemm_conv<0><<<gd, 256, 0, stream>>>(Wdv, xbuf, (long)DMODEL * THW, THW, 0,
                                             dvb, (long)FF * DP, DP, 0, FF, DP, DMODEL);

        // obs Q/K/V: M=128, N=2688, K=512
        dim3 go(tb_obs, BATCH);
        gemm_conv<0><<<go, 256, 0, stream>>>(Woq, xbuf, (long)DMODEL * THW, THW, DP,
                                             oqb, (long)FF * OP, OP, 0, FF, OP, DMODEL);
        gemm_conv<0><<<go, 256, 0, stream>>>(Wok, xbuf, (long)DMODEL * THW, THW, DP,
                                             okb, (long)FF * OP, OP, 0, FF, OP, DMODEL);
        gemm_conv<0><<<go, 256, 0, stream>>>(Wov, xbuf, (long)DMODEL * THW, THW, DP,
                                             ovb, (long)FF * OP, OP, 0, FF, OP, DMODEL);

        demo_attn<<<dim3(DP / 16, NHEAD, BATCH), 256, 0, stream>>>(dkb, dqb, dvb, dvatt);
        obs_attn<<<dim3(OBS_T, NHEAD, BATCH), 384, 0, stream>>>(dkb, dvatt, oqb, okb, ovb, ovatt);

        // W_o + residual relu into xbuf obs region: M=512, N=2688, K=128
        dim3 gw(tb_wo, BATCH);
        gemm_conv<1><<<gw, 256, 0, stream>>>(Woo, ovatt, (long)FF * OP, OP, 0,
                                             xbuf, (long)DMODEL * THW, THW, DP,
                                             DMODEL, OP, FF);

        bn_stats<<<DMODEL, 256, 0, stream>>>(xbuf, meanb, rstdb);
        float* xo = (L == 1) ? (float*)d_out : xbuf;
        bn_apply<<<(int)((XSZ + 255) / 256), 256, 0, stream>>>(xbuf, xo, meanb, rstdb,
                                                               gamma + (long)L * DMODEL,
                                                               beta + (long)L * DMODEL);
    }
}